// REModel_72438918414865
// MI455X (gfx1250) — compile-verified
//
#include <hip/hip_runtime.h>
#include <math.h>

// Problem constants from the reference
#define B_ 256
#define S_ 512
#define H_ 1024
#define L_ 30
#define SCHUNK 8          // S split into 8 chunks of 64 rows for the span stage

typedef float v2f __attribute__((ext_vector_type(2)));
typedef float v8f __attribute__((ext_vector_type(8)));

// ---------------------------------------------------------------------------
// Kernel 1a: deterministic partial span sums.
//   grid (B, SCHUNK); block (b, ry) accumulates rows [ry*64,(ry+1)*64) that
//   intersect each span and writes a per-chunk partial (no atomics ->
//   bitwise-deterministic across replays). 2048 blocks saturate HBM on the
//   dominant ~220 MB of x-span traffic.
// ---------------------------------------------------------------------------
__global__ __launch_bounds__(256) void re_span_partial(const float* __restrict__ x,
                                                       const int* __restrict__ eidx,
                                                       float* __restrict__ pSub,
                                                       float* __restrict__ pObj) {
    const int b   = blockIdx.x;
    const int ry  = blockIdx.y;
    const int col = threadIdx.x * 4;
    const int r0  = ry * (S_ / SCHUNK);
    const int r1  = r0 + (S_ / SCHUNK);

    const int i0 = eidx[b * 4 + 0];
    const int i1 = eidx[b * 4 + 1];
    const int i2 = eidx[b * 4 + 2];
    const int i3 = eidx[b * 4 + 3];

    const float* xb = x + (size_t)b * S_ * H_;

    float4 s = make_float4(0.f, 0.f, 0.f, 0.f);
    {
        const int lo = max(i0, r0), hi = min(i1, r1);
        for (int r = lo; r < hi; ++r) {
            float4 v = *(const float4*)(xb + (size_t)r * H_ + col);
            s.x += v.x; s.y += v.y; s.z += v.z; s.w += v.w;
        }
    }
    float4 o = make_float4(0.f, 0.f, 0.f, 0.f);
    {
        const int lo = max(i2, r0), hi = min(i3, r1);
        for (int r = lo; r < hi; ++r) {
            float4 v = *(const float4*)(xb + (size_t)r * H_ + col);
            o.x += v.x; o.y += v.y; o.z += v.z; o.w += v.w;
        }
    }
    *(float4*)(pSub + ((size_t)b * SCHUNK + ry) * H_ + col) = s;
    *(float4*)(pObj + ((size_t)b * SCHUNK + ry) * H_ + col) = o;
}

// ---------------------------------------------------------------------------
// Kernel 1b: finalize features (fixed-order partial reduction -> mean -> tanh)
//   feats[b, 0:H ] = tanh(x[b,0,:])
//   feats[b, H:2H] = tanh(subMean); feats[b, 2H:3H] = tanh(objMean)
// ---------------------------------------------------------------------------
__global__ __launch_bounds__(256) void re_finalize(const float* __restrict__ x,
                                                   const int* __restrict__ eidx,
                                                   const float* __restrict__ pSub,
                                                   const float* __restrict__ pObj,
                                                   float* __restrict__ feats) {
    const int b   = blockIdx.x;
    const int col = threadIdx.x * 4;

    const int i0 = eidx[b * 4 + 0];
    const int i1 = eidx[b * 4 + 1];
    const int i2 = eidx[b * 4 + 2];
    const int i3 = eidx[b * 4 + 3];
    const float invs = 1.0f / fmaxf((float)(i1 - i0), 1.0f);
    const float invo = 1.0f / fmaxf((float)(i3 - i2), 1.0f);

    float4 s = make_float4(0.f, 0.f, 0.f, 0.f);
    float4 o = make_float4(0.f, 0.f, 0.f, 0.f);
#pragma unroll
    for (int c = 0; c < SCHUNK; ++c) {
        float4 ps = *(const float4*)(pSub + ((size_t)b * SCHUNK + c) * H_ + col);
        float4 po = *(const float4*)(pObj + ((size_t)b * SCHUNK + c) * H_ + col);
        s.x += ps.x; s.y += ps.y; s.z += ps.z; s.w += ps.w;
        o.x += po.x; o.y += po.y; o.z += po.z; o.w += po.w;
    }
    float4 c0 = *(const float4*)(x + (size_t)b * S_ * H_ + col);

    float* fb = feats + (size_t)b * (3 * H_);
    float4 t;
    t.x = tanhf(c0.x); t.y = tanhf(c0.y); t.z = tanhf(c0.z); t.w = tanhf(c0.w);
    *(float4*)(fb + col) = t;
    t.x = tanhf(s.x * invs); t.y = tanhf(s.y * invs);
    t.z = tanhf(s.z * invs); t.w = tanhf(s.w * invs);
    *(float4*)(fb + H_ + col) = t;
    t.x = tanhf(o.x * invo); t.y = tanhf(o.y * invo);
    t.z = tanhf(o.z * invo); t.w = tanhf(o.w * invo);
    *(float4*)(fb + 2 * H_ + col) = t;
}

// ---------------------------------------------------------------------------
// Kernel 2: C[M,N] = A[M,K] @ W[K,N] + bias[N], f32 via V_WMMA_F32_16X16X4_F32.
//
// Block: 256 threads = 8 wave32s -> 64x64 tile; wave (mt,np) owns 16x32.
// Software-pipelined: register-prefetch next 16-K chunk while WMMAs consume
// the current double-buffered LDS chunk (one barrier per chunk).
//
// WMMA f32 16x16x4 lane layout (ISA 7.12.2):
//   A operand: lane l -> M=l&15 ; VGPR v -> K = kk + 2*(l>>4) + v
//   B operand: lane l -> N=l&15 ; VGPR v -> K = kk + 2*(l>>4) + v
//   C/D:       lane l -> N=l&15 ; VGPR j -> M = j + 8*(l>>4)
//
// LDS:
//   ldsA[2][64][20] row-major   : K-pair contiguous -> aligned ds_load_b64;
//     read banks lm*20 mod 64 distinct; halves differ by 2 (mod-4 disjoint).
//   ldsB[2][64][18] COLUMN-major ([n][k]): each B operand is one aligned
//     ds_load_b64 (k even -> 8B aligned); banks 18*lm distinct, halves
//     can collide (18Δ ≡ 2 mod 64 has solution Δ = -7).
// K multiple of 16 (1024, 3072); M multiple of 64. FULL_N=false adds the
// N-guard only for the N=30 head, keeping the big GEMMs branch-free.
// ---------------------------------------------------------------------------
template <bool FULL_N>
__global__ __launch_bounds__(256) void wmma_gemm_bias(
    const float* __restrict__ A, int lda,
    const float* __restrict__ W, int ldw,
    const float* __restrict__ bias,
    float* __restrict__ C, int ldc,
    int K, int N) {

    __shared__ float ldsA[2][64][20];
    __shared__ float ldsB[2][64][18];   // [n][k], column-major

    const int t    = threadIdx.x;
    const int lane = t & 31;
    const int wave = t >> 5;
    const int lm   = lane & 15;
    const int half = lane >> 4;
    const int mt   = wave & 3;   // 4 M strips of 16
    const int np   = wave >> 2;  // 2 N strips of 32

    const int rowBase = blockIdx.x * 64;
    const int colBase = blockIdx.y * 64;

    // cooperative staging assignments (coalesced float4 global loads)
    const int arow = t >> 2;        // 0..63
    const int akc  = (t & 3) * 4;   // 0,4,8,12
    const int brow = t >> 4;        // 0..15 (K within chunk)
    const int bc   = (t & 15) * 4;  // 0..60 (N within tile)

    const float* aPtr = A + (size_t)(rowBase + arow) * lda + akc;

    float4 av, bv;
    // prologue: fetch + stage chunk 0
    av = *(const float4*)(aPtr);
    {
        const float* wr = W + (size_t)brow * ldw;
        if (FULL_N) {
            bv = *(const float4*)(wr + colBase + bc);
        } else {
            const int n4 = colBase + bc;
            bv.x = (n4 + 0 < N) ? wr[n4 + 0] : 0.f;
            bv.y = (n4 + 1 < N) ? wr[n4 + 1] : 0.f;
            bv.z = (n4 + 2 < N) ? wr[n4 + 2] : 0.f;
            bv.w = (n4 + 3 < N) ? wr[n4 + 3] : 0.f;
        }
    }
    *(float4*)&ldsA[0][arow][akc] = av;
    ldsB[0][bc + 0][brow] = bv.x;
    ldsB[0][bc + 1][brow] = bv.y;
    ldsB[0][bc + 2][brow] = bv.z;
    ldsB[0][bc + 3][brow] = bv.w;
    __syncthreads();

    v8f acc0 = {};
    v8f acc1 = {};
    const int nchunks = K >> 4;
    const int n0l = np * 32 + lm;

    for (int c = 0; c < nchunks; ++c) {
        const int buf  = c & 1;
        const bool more = (c + 1) < nchunks;
        if (more) {
            const int kn = (c + 1) * 16;
            av = *(const float4*)(aPtr + kn);
            const float* wr = W + (size_t)(kn + brow) * ldw;
            if (FULL_N) {
                bv = *(const float4*)(wr + colBase + bc);
            } else {
                const int n4 = colBase + bc;
                bv.x = (n4 + 0 < N) ? wr[n4 + 0] : 0.f;
                bv.y = (n4 + 1 < N) ? wr[n4 + 1] : 0.f;
                bv.z = (n4 + 2 < N) ? wr[n4 + 2] : 0.f;
                bv.w = (n4 + 3 < N) ? wr[n4 + 3] : 0.f;
            }
        }

        const float* As  = &ldsA[buf][mt * 16 + lm][0];
        const float* Bs0 = &ldsB[buf][n0l][0];
        const float* Bs1 = &ldsB[buf][n0l + 16][0];
#pragma unroll
        for (int kk = 0; kk < 16; kk += 4) {
            const int ka = kk + 2 * half;
            v2f a  = *(const v2f*)(As + ka);
            v2f b0 = *(const v2f*)(Bs0 + ka);
            v2f b1 = *(const v2f*)(Bs1 + ka);
            acc0 = __builtin_amdgcn_wmma_f32_16x16x4_f32(
                false, a, false, b0, (short)0, acc0, false, false);
            acc1 = __builtin_amdgcn_wmma_f32_16x16x4_f32(
                false, a, false, b1, (short)0, acc1, false, false);
        }

        if (more) {
            const int nb = buf ^ 1;
            *(float4*)&ldsA[nb][arow][akc] = av;
            ldsB[nb][bc + 0][brow] = bv.x;
            ldsB[nb][bc + 1][brow] = bv.y;
            ldsB[nb][bc + 2][brow] = bv.z;
            ldsB[nb][bc + 3][brow] = bv.w;
            __syncthreads();
        }
    }

    // epilogue: bias add + store (N-guard only when needed)
    const int n0 = colBase + np * 32 + lm;
    const int n1 = n0 + 16;
    const float bz0 = (FULL_N || n0 < N) ? bias[n0] : 0.f;
    const float bz1 = (FULL_N || n1 < N) ? bias[n1] : 0.f;
#pragma unroll
    for (int j = 0; j < 8; ++j) {
        const int m = rowBase + mt * 16 + j + 8 * half;
        if (FULL_N || n0 < N) C[(size_t)m * ldc + n0] = acc0[j] + bz0;
        if (FULL_N || n1 < N) C[(size_t)m * ldc + n1] = acc1[j] + bz1;
    }
}

// ---------------------------------------------------------------------------
// Launch: span partials -> finalize feats -> block-diag GEMMs -> d1 -> d2 -> out
// Workspace layout (floats), with deliberate aliasing (partials die before Y
// is written; stream order guarantees safety):
//   [feats : B*3H] [region2 ...]
//   region2 as partials: pSub[B*8*H] | pObj[B*8*H]    (16 MB)
//   region2 as GEMM tmp: Y[B*3H] | h1[B*H] | h2[B*H]  ( 5 MB)
// Total ~20 MB.
// ---------------------------------------------------------------------------
extern "C" void kernel_launch(void* const* d_in, const int* in_sizes, int n_in,
                              void* d_out, int out_size, void* d_ws, size_t ws_size,
                              hipStream_t stream) {
    const float* x     = (const float*)d_in[0];
    const int*   eidx  = (const int*)d_in[1];
    const float* W_cls = (const float*)d_in[2];
    const float* b_cls = (const float*)d_in[3];
    const float* W_e1  = (const float*)d_in[4];
    const float* b_e1  = (const float*)d_in[5];
    const float* W_e2  = (const float*)d_in[6];
    const float* b_e2  = (const float*)d_in[7];
    const float* W_d1  = (const float*)d_in[8];
    const float* b_d1  = (const float*)d_in[9];
    const float* W_d2  = (const float*)d_in[10];
    const float* b_d2  = (const float*)d_in[11];
    const float* W_out = (const float*)d_in[12];
    const float* b_out = (const float*)d_in[13];
    float* out = (float*)d_out;

    float* feats   = (float*)d_ws;                       // B*3H
    float* region2 = feats + (size_t)B_ * 3 * H_;
    float* pSub    = region2;                            // B*SCHUNK*H
    float* pObj    = pSub + (size_t)B_ * SCHUNK * H_;    // B*SCHUNK*H
    float* Y       = region2;                            // B*3H (after partials die)
    float* h1      = Y + (size_t)B_ * 3 * H_;            // B*H
    float* h2      = h1 + (size_t)B_ * H_;               // B*H

    re_span_partial<<<dim3(B_, SCHUNK), dim3(256), 0, stream>>>(x, eidx, pSub, pObj);
    re_finalize<<<dim3(B_), dim3(256), 0, stream>>>(x, eidx, pSub, pObj, feats);

    const dim3 blk(256);
    const dim3 gHH(B_ / 64, H_ / 64);
    const dim3 gHL(B_ / 64, 1);

    // block-diagonal stage: Y = [x0 | e1 | e2]
    wmma_gemm_bias<true><<<gHH, blk, 0, stream>>>(feats + 0,      3 * H_, W_cls, H_, b_cls,
                                                  Y + 0,          3 * H_, H_, H_);
    wmma_gemm_bias<true><<<gHH, blk, 0, stream>>>(feats + H_,     3 * H_, W_e1,  H_, b_e1,
                                                  Y + H_,         3 * H_, H_, H_);
    wmma_gemm_bias<true><<<gHH, blk, 0, stream>>>(feats + 2 * H_, 3 * H_, W_e2,  H_, b_e2,
                                                  Y + 2 * H_,     3 * H_, H_, H_);
    // h1 = Y @ W_d1 + b_d1   (K = 3H)
    wmma_gemm_bias<true><<<gHH, blk, 0, stream>>>(Y, 3 * H_, W_d1, H_, b_d1, h1, H_, 3 * H_, H_);
    // h2 = h1 @ W_d2 + b_d2
    wmma_gemm_bias<true><<<gHH, blk, 0, stream>>>(h1, H_, W_d2, H_, b_d2, h2, H_, H_, H_);
    // out = h2 @ W_out + b_out  (N = 30, guarded)
    wmma_gemm_bias<false><<<gHL, blk, 0, stream>>>(h2, H_, W_out, L_, b_out, out, L_, H_, L_);
}